// HungarianMatcher_26173530702079
// MI455X (gfx1250) — compile-verified
//
#include <hip/hip_runtime.h>
#include <hip/hip_bf16.h>
#include <stdint.h>

// ---------------------------------------------------------------------------
// DETR Hungarian matcher cost matrix on gfx1250 (MI455X).
//   out[q, t] = 5 * L1(box_q, box_t) + delta[q, id_t] - 2 * GIoU(q, t)
// Kernel 1: per-query softmax + focal-cost table delta[14400, 128] -> ws
// Kernel 2: pairwise tile kernel:
//   - async global->LDS staging (global_load_async_to_lds_b128 + s_wait_asynccnt)
//   - non-temporal b128 output stores (184 MB stream must not sweep 192 MB L2)
// ---------------------------------------------------------------------------

#define CLS      128       // number of classes (fixed by problem)
#define TQ       32        // queries per block in pairwise kernel
#define NTHREADS 256       // threads per block in pairwise kernel
#define TPT      4         // targets per thread (float4 stores)
#define TT       (NTHREADS * TPT)   // 1024 targets per block

// native clang vector for nontemporal builtins (HIP float4 is a struct)
typedef float vf4 __attribute__((ext_vector_type(4)));

__device__ __forceinline__ unsigned lds_off(const void* p) {
    // Generic pointers into LDS carry the LDS byte offset in the low 32 bits
    // (the aperture calculation keeps only the low 32 bits).
    return (unsigned)(uintptr_t)p;
}

// ---------------------------------------------------------------------------
// Kernel 1: delta[q,c] = ALPHA*(1-p)^2*(-log(p+EPS)) - (1-ALPHA)*p^2*(-log(1-p+EPS))
// One wave32 per query; lane handles 4 classes (float4 in / float4 out).
// ---------------------------------------------------------------------------
__device__ __forceinline__ float focal_delta(float p) {
    const float ALPHA = 0.25f, EPS = 1e-8f;
    float omp = 1.0f - p;
    float pos = ALPHA * omp * omp * (-__logf(p + EPS));
    float neg = (1.0f - ALPHA) * p * p * (-__logf(omp + EPS));
    return pos - neg;
}

__global__ __launch_bounds__(256)
void focal_delta_kernel(const float* __restrict__ logits,  // [NQ, 128]
                        float* __restrict__ delta,         // [NQ, 128]
                        int NQ)
{
    int q    = (int)((blockIdx.x * blockDim.x + threadIdx.x) >> 5);
    int lane = threadIdx.x & 31;
    if (q >= NQ) return;

    // streamed, read-once: non-temporal load
    vf4 x = __builtin_nontemporal_load(
        (const vf4*)(logits + (size_t)q * CLS) + lane);

    float m = fmaxf(fmaxf(x.x, x.y), fmaxf(x.z, x.w));
    #pragma unroll
    for (int s = 16; s > 0; s >>= 1) m = fmaxf(m, __shfl_xor(m, s, 32));

    float e0 = __expf(x.x - m), e1 = __expf(x.y - m);
    float e2 = __expf(x.z - m), e3 = __expf(x.w - m);
    float sum = (e0 + e1) + (e2 + e3);
    #pragma unroll
    for (int s = 16; s > 0; s >>= 1) sum += __shfl_xor(sum, s, 32);
    float inv = 1.0f / sum;

    vf4 d;
    d.x = focal_delta(e0 * inv);
    d.y = focal_delta(e1 * inv);
    d.z = focal_delta(e2 * inv);
    d.w = focal_delta(e3 * inv);
    // delta table is re-read by kernel 2: keep regular-temporal store
    ((vf4*)(delta + (size_t)q * CLS))[lane] = d;
}

// ---------------------------------------------------------------------------
// Kernel 2: pairwise cost tile. blockIdx.x -> target tile, blockIdx.y -> query tile.
// Stages delta rows + query boxes into LDS with global_load_async_to_lds_b128
// (ASYNCcnt), then each thread owns 4 consecutive targets and loops 32 queries,
// emitting one non-temporal coalesced global_store_b128 per query.
// ---------------------------------------------------------------------------
__global__ __launch_bounds__(NTHREADS)
void pairwise_cost_kernel(const float* __restrict__ pred_boxes,  // [NQ, 4] cxcywh
                          const int*   __restrict__ tgt_ids,     // [T]
                          const float* __restrict__ tgt_bbox,    // [T, 4] cxcywh
                          const float* __restrict__ delta,       // [NQ, 128]
                          float* __restrict__ out,               // [NQ, T]
                          int NQ, int T)
{
    __shared__ __align__(16) float  sDelta[TQ * CLS];  // 16 KB focal table slice
    __shared__ float4 sQraw[TQ];                       // raw cxcywh
    __shared__ float4 sQder[TQ * 3];                   // [cxcywh][x0y0x1y1][area,-,-,-]

    const int tid   = threadIdx.x;
    const int qbase = blockIdx.y * TQ;
    const int tbase = blockIdx.x * TT;

    // ---- async stage: delta rows (TQ*CLS floats) + query boxes ----
    {
        const float* gdelta   = delta + (size_t)qbase * CLS;
        const unsigned availF = (unsigned)((NQ - qbase) * CLS);     // floats available
        const unsigned dbase  = lds_off(&sDelta[0]);
        #pragma unroll
        for (int j = 0; j < TQ * CLS / 4 / NTHREADS; ++j) {         // 4 iters
            unsigned v = (unsigned)(j * NTHREADS + tid);            // float4 index
            if (v * 4u < availF) {
                unsigned b = v * 16u;                               // byte offset
                asm volatile("global_load_async_to_lds_b128 %0, %1, %2"
                             :: "v"(dbase + b), "v"(b), "s"(gdelta)
                             : "memory");
            }
        }
        if (tid < TQ && (qbase + tid) < NQ) {
            const float* gq = pred_boxes + (size_t)qbase * 4;
            unsigned b = (unsigned)tid * 16u;
            asm volatile("global_load_async_to_lds_b128 %0, %1, %2"
                         :: "v"(lds_off(&sQraw[0]) + b), "v"(b), "s"(gq)
                         : "memory");
        }
        asm volatile("s_wait_asynccnt 0" ::: "memory");
    }
    __syncthreads();

    // ---- derive per-query xyxy + area once (32 threads) ----
    if (tid < TQ) {
        float4 b = sQraw[tid];
        float hw = 0.5f * b.z, hh = 0.5f * b.w;
        sQder[tid * 3 + 0] = b;
        sQder[tid * 3 + 1] = make_float4(b.x - hw, b.y - hh, b.x + hw, b.y + hh);
        sQder[tid * 3 + 2] = make_float4(b.z * b.w, 0.f, 0.f, 0.f);
    }
    __syncthreads();

    const int t0 = tbase + tid * TPT;
    if (t0 >= T) return;   // partial last tile (T % 4 == 0, vector all-or-nothing)

    // ---- per-target data in registers ----
    int4 ids = *(const int4*)(tgt_ids + t0);
    int  idv[TPT] = {ids.x, ids.y, ids.z, ids.w};
    float4 tb[TPT];
    float tx0[TPT], ty0[TPT], tx1[TPT], ty1[TPT], tarea[TPT];
    #pragma unroll
    for (int j = 0; j < TPT; ++j) {
        tb[j] = *(const float4*)(tgt_bbox + (size_t)(t0 + j) * 4);
        float hw = 0.5f * tb[j].z, hh = 0.5f * tb[j].w;
        tx0[j] = tb[j].x - hw;  ty0[j] = tb[j].y - hh;
        tx1[j] = tb[j].x + hw;  ty1[j] = tb[j].y + hh;
        tarea[j] = tb[j].z * tb[j].w;
    }

    const int qmax = min(TQ, NQ - qbase);
    for (int qq = 0; qq < qmax; ++qq) {
        float4 qb = sQder[qq * 3 + 0];   // cx,cy,w,h (LDS broadcast)
        float4 qx = sQder[qq * 3 + 1];   // x0,y0,x1,y1
        float  qa = sQder[qq * 3 + 2].x; // area

        float r[TPT];
        #pragma unroll
        for (int j = 0; j < TPT; ++j) {
            // L1 on cxcywh
            float l1 = fabsf(qb.x - tb[j].x) + fabsf(qb.y - tb[j].y)
                     + fabsf(qb.z - tb[j].z) + fabsf(qb.w - tb[j].w);
            // GIoU
            float ix0 = fmaxf(qx.x, tx0[j]), iy0 = fmaxf(qx.y, ty0[j]);
            float ix1 = fminf(qx.z, tx1[j]), iy1 = fminf(qx.w, ty1[j]);
            float iw  = fmaxf(ix1 - ix0, 0.f), ih = fmaxf(iy1 - iy0, 0.f);
            float inter = iw * ih;
            float uni   = qa + tarea[j] - inter;
            float iou   = inter / uni;
            float ex0 = fminf(qx.x, tx0[j]), ey0 = fminf(qx.y, ty0[j]);
            float ex1 = fmaxf(qx.z, tx1[j]), ey1 = fmaxf(qx.w, ty1[j]);
            float ew  = fmaxf(ex1 - ex0, 0.f), eh = fmaxf(ey1 - ey0, 0.f);
            float ea  = ew * eh;
            float giou = iou - (ea - uni) / ea;
            // focal class gather from LDS
            float cl = sDelta[qq * CLS + idv[j]];
            r[j] = 5.0f * l1 + cl - 2.0f * giou;
        }
        vf4 res = {r[0], r[1], r[2], r[3]};
        // write-once 184 MB stream: non-temporal b128 store, keep L2 for the
        // reused delta table / target boxes
        __builtin_nontemporal_store(res,
            (vf4*)(out + (size_t)(qbase + qq) * T + t0));
    }
}

// ---------------------------------------------------------------------------
extern "C" void kernel_launch(void* const* d_in, const int* in_sizes, int n_in,
                              void* d_out, int out_size, void* d_ws, size_t ws_size,
                              hipStream_t stream) {
    const float* pred_logits = (const float*)d_in[0];  // [bs*Q, 128]
    const float* pred_boxes  = (const float*)d_in[1];  // [bs*Q, 4]
    const int*   tgt_ids     = (const int*)d_in[2];    // [T]
    const float* tgt_bbox    = (const float*)d_in[3];  // [T, 4]
    float*       out         = (float*)d_out;          // [bs*Q, T]
    float*       ws_delta    = (float*)d_ws;           // [bs*Q, 128] scratch

    const int NQ = in_sizes[0] / CLS;   // 14400
    const int T  = in_sizes[2];         // 3200

    // Kernel 1: 8 queries (waves) per 256-thread block
    int blocks1 = (NQ + 7) / 8;
    focal_delta_kernel<<<blocks1, 256, 0, stream>>>(pred_logits, ws_delta, NQ);

    // Kernel 2: tiles of 32 queries x 1024 targets
    dim3 grid2((T + TT - 1) / TT, (NQ + TQ - 1) / TQ);
    pairwise_cost_kernel<<<grid2, NTHREADS, 0, stream>>>(
        pred_boxes, tgt_ids, tgt_bbox, ws_delta, out, NQ, T);
}